// SPR_RGCN_88648124990120
// MI455X (gfx1250) — compile-verified
//
#include <hip/hip_runtime.h>

// ---------------------------------------------------------------------------
// RGCN (2 layers, mean aggregation per relation) + mean pool + linear head.
// f32 everywhere; dense node transforms use V_WMMA_F32_16X16X4_F32 so the
// reference numerics are preserved exactly while the matrix pipe does the GEMM.
// ---------------------------------------------------------------------------

typedef __attribute__((ext_vector_type(2))) float v2f;
typedef __attribute__((ext_vector_type(8))) float v8f;

#define H 128           // hidden/embed width
#define R 3             // relations
#define NCLS 10

__device__ __forceinline__ void atomAddF(float* p, float v) {
    (void)__hip_atomic_fetch_add(p, v, __ATOMIC_RELAXED, __HIP_MEMORY_SCOPE_AGENT);
}

// ---------------- zero ----------------
__global__ void k_zero(float* __restrict__ p, long long n) {
    long long i = (long long)blockIdx.x * blockDim.x + threadIdx.x;
    if (i < n) p[i] = 0.0f;
}

// ---------------- embedding gather: h[n,:] = embed[x[n],:] ----------------
__global__ void k_embed(const int* __restrict__ x, const float* __restrict__ embed,
                        float* __restrict__ h, int N) {
    long long idx = (long long)blockIdx.x * blockDim.x + threadIdx.x;
    long long total = (long long)N * (H / 4);
    if (idx >= total) return;
    int n = (int)(idx >> 5);            // H/4 == 32 chunks per node
    int c = ((int)idx & 31) << 2;
    const float4 v = *(const float4*)(embed + (size_t)x[n] * H + c);
    *(float4*)(h + (size_t)n * H + c) = v;
}

// ---------------- per-(dst,rel) edge counts ----------------
__global__ void k_count(const int* __restrict__ dstI, const int* __restrict__ et,
                        float* __restrict__ cnt, int E) {
    int e = blockIdx.x * blockDim.x + threadIdx.x;
    if (e >= E) return;
    atomAddF(&cnt[(size_t)dstI[e] * R + et[e]], 1.0f);
}

// ---------------- WMMA f32 GEMM: T[w] = A @ W_w  (W_w 128x128) ----------------
// blockIdx.x = row tile (16 rows), blockIdx.y = weight select (0..2 rel, 3 root)
// 256 threads = 8 waves; wave w owns output columns [16w, 16w+16).
__global__ __launch_bounds__(256)
void k_gemm(const float* __restrict__ A, const float* __restrict__ Wrel,
            const float* __restrict__ Wroot, float* __restrict__ T, int N) {
    __shared__ float Ash[16 * 132];     // stride 132 -> conflict-free column reads

    const int rt = blockIdx.x;
    const int w  = blockIdx.y;
    const float* __restrict__ W = (w < R) ? (Wrel + (size_t)w * H * H) : Wroot;
    float* __restrict__ Out = T + (size_t)w * N * H;

    // Cooperative A-tile load: 16x128 floats, 8 floats per thread.
    {
        int flat = threadIdx.x * 8;
        int row = flat >> 7, col = flat & 127;
        int gr = rt * 16 + row;
        float4 v0 = make_float4(0.f, 0.f, 0.f, 0.f), v1 = v0;
        if (gr < N) {
            const float* s = A + (size_t)gr * H + col;
            v0 = *(const float4*)s;
            v1 = *(const float4*)(s + 4);
        }
        *(float4*)&Ash[row * 132 + col]     = v0;
        *(float4*)&Ash[row * 132 + col + 4] = v1;
    }
    __syncthreads();

    const int lane    = threadIdx.x & 31;
    const int wave    = threadIdx.x >> 5;
    const int colbase = wave * 16;
    const int m       = lane & 15;      // row (A) / column (B) within tile
    const int half    = lane >> 4;      // K-pair select per ISA f32 layouts

    v8f c = {};
    #pragma unroll 4
    for (int k = 0; k < 32; ++k) {      // K = 128 in steps of 4
        const int kx = 4 * k + 2 * half;
        v2f a, b;
        a.x = Ash[m * 132 + kx];
        a.y = Ash[m * 132 + kx + 1];
        b.x = W[(size_t)kx * H + colbase + m];
        b.y = W[(size_t)(kx + 1) * H + colbase + m];
        c = __builtin_amdgcn_wmma_f32_16x16x4_f32(
                /*neg_a=*/false, a, /*neg_b=*/false, b,
                /*c_mod=*/(short)0, c, /*reuse_a=*/false, /*reuse_b=*/false);
    }

    // Epilogue: C/D layout -> VGPR v holds row (v + 8*half), column = lane%16.
    // Uniform full-tile fast path: one scalar branch, 8 immediate-offset stores
    // (store clause), instead of per-row exec-mask juggling.
    const int gcol = colbase + m;
    if (rt * 16 + 16 <= N) {
        float* outp = Out + ((size_t)(rt * 16 + 8 * half)) * H + gcol;
        #pragma unroll
        for (int v = 0; v < 8; ++v)
            outp[(size_t)v * H] = c[v];
    } else {
        #pragma unroll
        for (int v = 0; v < 8; ++v) {
            int grow = rt * 16 + v + 8 * half;
            if (grow < N) Out[(size_t)grow * H + gcol] = c[v];
        }
    }
}

// ---------------- edge scatter: sums[(dst,rel),:] += t[rel, src, :] ----------------
__global__ void k_scatter(const int* __restrict__ srcI, const int* __restrict__ dstI,
                          const int* __restrict__ et, const float* __restrict__ T,
                          float* __restrict__ sums, int E, int N) {
    long long idx = (long long)blockIdx.x * blockDim.x + threadIdx.x;
    long long total = (long long)E * (H / 4);
    if (idx >= total) return;
    int e = (int)(idx >> 5);
    int c = ((int)idx & 31) << 2;
    const int r = et[e], s = srcI[e], d = dstI[e];
    const float4 v = *(const float4*)(T + ((size_t)r * N + s) * H + c);
    float* p = sums + ((size_t)d * R + r) * H + c;
    atomAddF(p + 0, v.x); atomAddF(p + 1, v.y);
    atomAddF(p + 2, v.z); atomAddF(p + 3, v.w);
}

// ---------------- combine: relu(sum_r sums/max(cnt,1) + root + bias) ----------------
__global__ void k_combine(const float* __restrict__ sums, const float* __restrict__ cnt,
                          const float* __restrict__ Troot, const float* __restrict__ bias,
                          float* __restrict__ hout, int N) {
    long long idx = (long long)blockIdx.x * blockDim.x + threadIdx.x;
    long long total = (long long)N * (H / 4);
    if (idx >= total) return;
    int n = (int)(idx >> 5);
    int c = ((int)idx & 31) << 2;
    float4 acc = make_float4(0.f, 0.f, 0.f, 0.f);
    #pragma unroll
    for (int r = 0; r < R; ++r) {
        const float inv = 1.0f / fmaxf(cnt[(size_t)n * R + r], 1.0f);
        const float4 s = *(const float4*)(sums + ((size_t)n * R + r) * H + c);
        acc.x += s.x * inv; acc.y += s.y * inv;
        acc.z += s.z * inv; acc.w += s.w * inv;
    }
    const float4 rt = *(const float4*)(Troot + (size_t)n * H + c);
    const float4 bb = *(const float4*)(bias + c);
    float4 o;
    o.x = fmaxf(acc.x + rt.x + bb.x, 0.f);
    o.y = fmaxf(acc.y + rt.y + bb.y, 0.f);
    o.z = fmaxf(acc.z + rt.z + bb.z, 0.f);
    o.w = fmaxf(acc.w + rt.w + bb.w, 0.f);
    *(float4*)(hout + (size_t)n * H + c) = o;
}

// ---------------- graph pooling ----------------
__global__ void k_poolcnt(const int* __restrict__ batch, float* __restrict__ pcnt, int N) {
    int n = blockIdx.x * blockDim.x + threadIdx.x;
    if (n >= N) return;
    atomAddF(&pcnt[batch[n]], 1.0f);
}

__global__ void k_pooladd(const int* __restrict__ batch, const float* __restrict__ h,
                          float* __restrict__ pool, int N) {
    long long idx = (long long)blockIdx.x * blockDim.x + threadIdx.x;
    long long total = (long long)N * (H / 4);
    if (idx >= total) return;
    int n = (int)(idx >> 5);
    int c = ((int)idx & 31) << 2;
    const int g = batch[n];
    const float4 v = *(const float4*)(h + (size_t)n * H + c);
    float* p = pool + (size_t)g * H + c;
    atomAddF(p + 0, v.x); atomAddF(p + 1, v.y);
    atomAddF(p + 2, v.z); atomAddF(p + 3, v.w);
}

// ---------------- classifier head: out[g,c] = mean_g . Wlin[:,c] + blin[c] ----------------
__global__ void k_final(const float* __restrict__ pool, const float* __restrict__ pcnt,
                        const float* __restrict__ Wlin, const float* __restrict__ blin,
                        float* __restrict__ out, int G) {
    int idx = blockIdx.x * blockDim.x + threadIdx.x;
    if (idx >= G * NCLS) return;
    const int g = idx / NCLS, cls = idx % NCLS;
    const float inv = 1.0f / fmaxf(pcnt[g], 1.0f);
    float acc = blin[cls];
    #pragma unroll 8
    for (int i = 0; i < H; ++i)
        acc += pool[(size_t)g * H + i] * inv * Wlin[(size_t)i * NCLS + cls];
    out[(size_t)g * NCLS + cls] = acc;
}

// ---------------------------------------------------------------------------
extern "C" void kernel_launch(void* const* d_in, const int* in_sizes, int n_in,
                              void* d_out, int out_size, void* d_ws, size_t ws_size,
                              hipStream_t stream) {
    const int*   x      = (const int*)d_in[0];
    const int*   eidx   = (const int*)d_in[1];
    const int*   etype  = (const int*)d_in[2];
    const int*   batch  = (const int*)d_in[3];
    /* d_in[4] = num_graphs scalar on device; derived from out_size instead */
    const float* embed  = (const float*)d_in[5];
    const float* W1rel  = (const float*)d_in[6];
    const float* W1root = (const float*)d_in[7];
    const float* b1     = (const float*)d_in[8];
    const float* W2rel  = (const float*)d_in[9];
    const float* W2root = (const float*)d_in[10];
    const float* b2     = (const float*)d_in[11];
    const float* Wlin   = (const float*)d_in[12];
    const float* blin   = (const float*)d_in[13];
    float* out = (float*)d_out;

    const int N = in_sizes[0];          // nodes
    const int E = in_sizes[2];          // edges
    const int G = out_size / NCLS;      // graphs
    const int* srcI = eidx;
    const int* dstI = eidx + E;

    // workspace carve-up (f32)
    float* t    = (float*)d_ws;                       // [4][N][H]  (3 rel + root)
    float* sums = t    + (size_t)4 * N * H;           // [N*R][H]
    float* cnt  = sums + (size_t)R * N * H;           // [N*R]
    float* hA   = cnt  + (size_t)R * N;               // [N][H]
    float* hB   = hA   + (size_t)N * H;               // [N][H]
    float* pool = hB   + (size_t)N * H;               // [G][H]
    float* pcnt = pool + (size_t)G * H;               // [G]

    const int TPB = 256;
    const long long nh4   = (long long)N * (H / 4);
    const long long eh4   = (long long)E * (H / 4);
    const int nhBlocks    = (int)((nh4 + TPB - 1) / TPB);
    const int ehBlocks    = (int)((eh4 + TPB - 1) / TPB);
    const int rowTiles    = (N + 15) / 16;

    const long long zSumsCnt = (long long)R * N * H + (long long)R * N;
    const long long zSums    = (long long)R * N * H;
    const long long zPool    = (long long)G * H + G;

    // ---- layer 1 ----
    k_zero<<<(int)((zSumsCnt + TPB - 1) / TPB), TPB, 0, stream>>>(sums, zSumsCnt);
    k_embed<<<nhBlocks, TPB, 0, stream>>>(x, embed, hA, N);
    k_count<<<(E + TPB - 1) / TPB, TPB, 0, stream>>>(dstI, etype, cnt, E);
    k_gemm<<<dim3(rowTiles, 4), TPB, 0, stream>>>(hA, W1rel, W1root, t, N);
    k_scatter<<<ehBlocks, TPB, 0, stream>>>(srcI, dstI, etype, t, sums, E, N);
    k_combine<<<nhBlocks, TPB, 0, stream>>>(sums, cnt, t + (size_t)3 * N * H, b1, hB, N);

    // ---- layer 2 (counts reused; only sums re-zeroed) ----
    k_zero<<<(int)((zSums + TPB - 1) / TPB), TPB, 0, stream>>>(sums, zSums);
    k_gemm<<<dim3(rowTiles, 4), TPB, 0, stream>>>(hB, W2rel, W2root, t, N);
    k_scatter<<<ehBlocks, TPB, 0, stream>>>(srcI, dstI, etype, t, sums, E, N);
    k_combine<<<nhBlocks, TPB, 0, stream>>>(sums, cnt, t + (size_t)3 * N * H, b2, hA, N);

    // ---- pool + head ----
    k_zero<<<(int)((zPool + TPB - 1) / TPB), TPB, 0, stream>>>(pool, zPool);
    k_poolcnt<<<(N + TPB - 1) / TPB, TPB, 0, stream>>>(batch, pcnt, N);
    k_pooladd<<<nhBlocks, TPB, 0, stream>>>(batch, hA, pool, N);
    k_final<<<(G * NCLS + TPB - 1) / TPB, TPB, 0, stream>>>(pool, pcnt, Wlin, blin, out, G);
}